// SymmetricContraction_57166014710102
// MI455X (gfx1250) — compile-verified
//
#include <hip/hip_runtime.h>
#include <hip/hip_bf16.h>

// ---------------- problem constants ----------------
#define BB   512
#define CC   128
#define II   9
#define EE   10
#define MM   3
#define NN   (BB * CC)          // 65536 "columns" (b,c)
#define K3S  7
#define K3V  10
// padded GEMM dims
#define MS_PAD 96               // 81 -> 96 rows (scalar U3)
#define KS_PAD 64               // 63 -> 64
#define MV_PAD 256              // 243 -> 256 rows (vector U3)
#define KV_PAD 96               // 90 -> 96

typedef float v2f __attribute__((ext_vector_type(2)));
typedef float v8f __attribute__((ext_vector_type(8)));

// ---------------- workspace layout (bytes) ----------------
// z tensors are stored as K-row PAIRS: zp[(kk>>1)*NN*2 + n*2 + (kk&1)]
// so one lane's two K values per WMMA step are a single aligned b64.
#define OFF_ELEM  0                         // 512 * 4            = 2 KB
#define OFF_U3SF  4096                      // 96*64*4            = 24576
#define OFF_U3VF  (OFF_U3SF + 24576)        // 256*96*4           = 98304
#define OFF_Z3S   131072                    // 64*65536*4         = 16777216
#define OFF_Z3V   (OFF_Z3S + 16777216)      // 96*65536*4         = 25165824
#define WS_NEEDED (OFF_Z3V + 25165824)      // ~42 MB

// ---------------- kernel 0: one-hot -> element index ----------------
__global__ void sc_elem_kernel(const float* __restrict__ y, int* __restrict__ elem) {
    int b = blockIdx.x * blockDim.x + threadIdx.x;
    if (b < BB) {
        int e = 0;
        for (int j = 0; j < EE; ++j)
            if (y[b * EE + j] > 0.5f) e = j;
        elem[b] = e;
    }
}

// ---------------- kernel 1: zero-padded flattened U3 (shared WMMA A operands) ----------------
__global__ void sc_build_a_kernel(const float* __restrict__ U3s, const float* __restrict__ U3v,
                                  float* __restrict__ U3s_flat, float* __restrict__ U3v_flat) {
    int t = blockIdx.x * blockDim.x + threadIdx.x;
    if (t < MS_PAD * KS_PAD) {
        int row = t >> 6, col = t & 63;            // [96][64]
        float v = 0.f;
        if (row < 81 && col < 63) {
            int i = row / 9, j = row % 9, a = col / K3S, k = col % K3S;
            v = U3s[((i * 9 + j) * 9 + a) * K3S + k];
        }
        U3s_flat[t] = v;
    } else {
        int t2 = t - MS_PAD * KS_PAD;
        if (t2 < MV_PAD * KV_PAD) {
            int row = t2 / KV_PAD, col = t2 % KV_PAD;  // [256][96]
            float v = 0.f;
            if (row < 243 && col < 90) {
                int m = row / 81, rem = row % 81, i = rem / 9, j = rem % 9;
                int a = col / K3V, k = col % K3V;
                v = U3v[(((m * 9 + i) * 9 + j) * 9 + a) * K3V + k];
            }
            U3v_flat[t2] = v;
        }
    }
}

// ---------------- kernel 2: z[(a,k), n] = x[n,a] * W3[elem(b), k, c], pair-packed ----------------
__global__ void sc_build_z_kernel(const float* __restrict__ x, const int* __restrict__ elem,
                                  const float* __restrict__ W3s, const float* __restrict__ W3v,
                                  float* __restrict__ z3s, float* __restrict__ z3v) {
    int n = blockIdx.x * blockDim.x + threadIdx.x;
    if (n >= NN) return;
    int b = n >> 7, c = n & 127;
    int e = elem[b];
    float xa[II];
#pragma unroll
    for (int a = 0; a < II; ++a) xa[a] = x[n * II + a];

    float vs[KS_PAD];
#pragma unroll
    for (int r = 0; r < KS_PAD; ++r) vs[r] = 0.f;
#pragma unroll
    for (int k = 0; k < K3S; ++k) {
        float w = W3s[(e * K3S + k) * CC + c];
#pragma unroll
        for (int a = 0; a < II; ++a) vs[a * K3S + k] = xa[a] * w;
    }
#pragma unroll
    for (int p = 0; p < KS_PAD / 2; ++p) {          // pair-packed b64 stores, coalesced in n
        v2f t; t[0] = vs[2 * p]; t[1] = vs[2 * p + 1];
        *(v2f*)(z3s + (size_t)(p * NN + n) * 2) = t;
    }

    float vv[KV_PAD];
#pragma unroll
    for (int r = 0; r < KV_PAD; ++r) vv[r] = 0.f;
#pragma unroll
    for (int k = 0; k < K3V; ++k) {
        float w = W3v[(e * K3V + k) * CC + c];
#pragma unroll
        for (int a = 0; a < II; ++a) vv[a * K3V + k] = xa[a] * w;
    }
#pragma unroll
    for (int p = 0; p < KV_PAD / 2; ++p) {
        v2f t; t[0] = vv[2 * p]; t[1] = vv[2 * p + 1];
        *(v2f*)(z3v + (size_t)(p * NN + n) * 2) = t;
    }
}

// ---------------- kernel 3: WMMA GEMMs + nu=2 / nu=1 finalize ----------------
// One wave32 per 16-column tile of n = b*C + c.
__global__ __launch_bounds__(32) void sc_main_kernel(
    const float* __restrict__ x, const int* __restrict__ elem,
    const float* __restrict__ z3s, const float* __restrict__ z3v,
    const float* __restrict__ U3s_flat, const float* __restrict__ U3v_flat,
    const float* __restrict__ U2s, const float* __restrict__ U2v,
    const float* __restrict__ U1s, const float* __restrict__ U1v,
    const float* __restrict__ W2s, const float* __restrict__ W2v,
    const float* __restrict__ W1s, const float* __restrict__ W1v,
    float* __restrict__ out)
{
    __shared__ float t2s[MS_PAD * 16];     // scalar GEMM result tile (rows x 16 cols)
    __shared__ float t2v[MV_PAD * 16];     // vector GEMM result tile
    __shared__ float u2s_l[9 * 9 * 3];
    __shared__ float u2v_l[3 * 9 * 9 * 3];
    __shared__ float u1s_l[9];
    __shared__ float u1v_l[3 * 9];

    const int lane = threadIdx.x;
    for (int i = lane; i < 243; i += 32) u2s_l[i] = U2s[i];
    for (int i = lane; i < 729; i += 32) u2v_l[i] = U2v[i];
    if (lane < 9)  u1s_l[lane] = U1s[lane];
    if (lane < 27) u1v_l[lane] = U1v[lane];

    const int half = lane >> 4;            // 0: lanes 0-15, 1: lanes 16-31
    const int lq   = lane & 15;
    const int n0   = blockIdx.x * 16;
    const int coln = n0 + lq;

    // ---- scalar GEMM: (96 x 64) x (64 x 16) via v_wmma_f32_16x16x4_f32 ----
    {
        v2f bp[16];                        // B panel in registers; one b64 per K-step
#pragma unroll
        for (int kt = 0; kt < 16; ++kt) {  // lane needs rows kt*4+half*2, +1 -> packed pair
            int p = kt * 2 + half;
            bp[kt] = *(const v2f*)(z3s + (size_t)(p * NN + coln) * 2);
        }
        for (int mt = 0; mt < 6; ++mt) {
            v8f acc = {};
            const float* arow = U3s_flat + (mt * 16 + lq) * KS_PAD + half * 2;
            __builtin_prefetch(arow + 16 * KS_PAD, 0, 1);   // next mt row -> global_prefetch
#pragma unroll
            for (int kt = 0; kt < 16; ++kt) {
                v2f a = *(const v2f*)(arow + kt * 4);
                acc = __builtin_amdgcn_wmma_f32_16x16x4_f32(
                    false, a, false, bp[kt], (short)0, acc, false, false);
            }
#pragma unroll
            for (int v = 0; v < 8; ++v) {  // D layout: lanes<16 M=v, lanes>=16 M=v+8
                int row = mt * 16 + v + half * 8;
                t2s[row * 16 + lq] = acc[v];
            }
        }
    }

    // ---- vector GEMM: (256 x 96) x (96 x 16) ----
    {
        v2f bpv[24];
#pragma unroll
        for (int kt = 0; kt < 24; ++kt) {
            int p = kt * 2 + half;
            bpv[kt] = *(const v2f*)(z3v + (size_t)(p * NN + coln) * 2);
        }
        for (int mt = 0; mt < 16; ++mt) {
            v8f acc = {};
            const float* arow = U3v_flat + (mt * 16 + lq) * KV_PAD + half * 2;
            if (mt < 15) __builtin_prefetch(arow + 16 * KV_PAD, 0, 1);
#pragma unroll
            for (int kt = 0; kt < 24; ++kt) {
                v2f a = *(const v2f*)(arow + kt * 4);
                acc = __builtin_amdgcn_wmma_f32_16x16x4_f32(
                    false, a, false, bpv[kt], (short)0, acc, false, false);
            }
#pragma unroll
            for (int v = 0; v < 8; ++v) {
                int row = mt * 16 + v + half * 8;
                t2v[row * 16 + lq] = acc[v];
            }
        }
    }

    __syncthreads();

    // ---- finalize: nu=2 and nu=1 stages per column ----
    const int col = lq;
    const int n = n0 + col;
    const int b = n >> 7, c = n & 127;
    const int e = elem[b];
    float xa[II];
#pragma unroll
    for (int a = 0; a < II; ++a) xa[a] = x[n * II + a];

    if (half == 0) {
        // scalar (0e) output for this column
        float w2[3];
#pragma unroll
        for (int k = 0; k < 3; ++k) w2[k] = W2s[(e * 3 + k) * CC + c];
        float t1[II];
#pragma unroll
        for (int i = 0; i < II; ++i) {
            float acc1 = 0.f;
#pragma unroll
            for (int a = 0; a < II; ++a) {
                float c2 = 0.f;
#pragma unroll
                for (int k = 0; k < 3; ++k) c2 += u2s_l[(i * 9 + a) * 3 + k] * w2[k];
                acc1 += (c2 + t2s[(i * 9 + a) * 16 + col]) * xa[a];
            }
            t1[i] = acc1;
        }
        float w1 = W1s[e * CC + c];
        float s = 0.f;
#pragma unroll
        for (int a = 0; a < II; ++a) s += (u1s_l[a] * w1 + t1[a]) * xa[a];
        out[b * 512 + c] = s;
    }

    // vector (1o) output: lanes<16 do m=0; lanes>=16 do m=1,2 (load balanced)
    {
        float w2v[3];
#pragma unroll
        for (int k = 0; k < 3; ++k) w2v[k] = W2v[(e * 3 + k) * CC + c];
        float w1v = W1v[e * CC + c];
        const int mstart = (half == 0) ? 0 : 1;
        const int mend   = (half == 0) ? 1 : 3;
        for (int m = mstart; m < mend; ++m) {
            float t1v[II];
#pragma unroll
            for (int i = 0; i < II; ++i) {
                float acc1 = 0.f;
#pragma unroll
                for (int a = 0; a < II; ++a) {
                    float c2 = 0.f;
#pragma unroll
                    for (int k = 0; k < 3; ++k)
                        c2 += u2v_l[((m * 9 + i) * 9 + a) * 3 + k] * w2v[k];
                    acc1 += (c2 + t2v[(m * 81 + i * 9 + a) * 16 + col]) * xa[a];
                }
                t1v[i] = acc1;
            }
            float v = 0.f;
#pragma unroll
            for (int a = 0; a < II; ++a) v += (u1v_l[m * 9 + a] * w1v + t1v[a]) * xa[a];
            out[b * 512 + CC + c * 3 + m] = v;
        }
    }
}

extern "C" void kernel_launch(void* const* d_in, const int* in_sizes, int n_in,
                              void* d_out, int out_size, void* d_ws, size_t ws_size,
                              hipStream_t stream) {
    (void)in_sizes; (void)n_in; (void)out_size;
    if (ws_size < (size_t)WS_NEEDED) return;   // workspace too small; cannot proceed

    const float* x   = (const float*)d_in[0];
    const float* y   = (const float*)d_in[1];
    const float* U1s = (const float*)d_in[2];
    const float* U2s = (const float*)d_in[3];
    const float* U3s = (const float*)d_in[4];
    const float* U1v = (const float*)d_in[5];
    const float* U2v = (const float*)d_in[6];
    const float* U3v = (const float*)d_in[7];
    const float* W1s = (const float*)d_in[8];
    const float* W2s = (const float*)d_in[9];
    const float* W3s = (const float*)d_in[10];
    const float* W1v = (const float*)d_in[11];
    const float* W2v = (const float*)d_in[12];
    const float* W3v = (const float*)d_in[13];
    float* out = (float*)d_out;

    char* ws = (char*)d_ws;
    int*   elem     = (int*)  (ws + OFF_ELEM);
    float* U3s_flat = (float*)(ws + OFF_U3SF);
    float* U3v_flat = (float*)(ws + OFF_U3VF);
    float* z3s      = (float*)(ws + OFF_Z3S);
    float* z3v      = (float*)(ws + OFF_Z3V);

    // 0) element index from one-hot
    sc_elem_kernel<<<2, 256, 0, stream>>>(y, elem);
    // 1) padded flattened U3 operands (shared WMMA A matrices)
    sc_build_a_kernel<<<(MS_PAD * KS_PAD + MV_PAD * KV_PAD + 255) / 256, 256, 0, stream>>>(
        U3s, U3v, U3s_flat, U3v_flat);
    // 2) z tensors (B operands), K-padded with zeros, pair-packed for b64 loads
    sc_build_z_kernel<<<NN / 256, 256, 0, stream>>>(x, elem, W3s, W3v, z3s, z3v);
    // 3) WMMA GEMMs + finalize; one wave32 per 16 columns
    sc_main_kernel<<<NN / 16, 32, 0, stream>>>(
        x, elem, z3s, z3v, U3s_flat, U3v_flat,
        U2s, U2v, U1s, U1v, W2s, W2v, W1s, W1v, out);
}